// general_model_53025666237063
// MI455X (gfx1250) — compile-verified
//
#include <hip/hip_runtime.h>

typedef float v2f __attribute__((ext_vector_type(2)));
typedef float v8f __attribute__((ext_vector_type(8)));

#define DIM      768
#define BM       128
#define BN       64
#define KC       32
#define NCHUNK   (DIM / KC)     // 24
#define ASTRIDE  (KC + 4)       // 36 floats (A tile row stride, padded)
#define BTSTRIDE (KC + 4)       // 36 floats (transposed B: Bt[n][k])
#define NTHREADS 256

// LDS byte offset of a __shared__ pointer (generic AS: low 32 bits = LDS offset)
#define LDS_OFF(p) ((unsigned)(unsigned long long)(p))

__device__ __forceinline__ void async_b128(unsigned ldsoff, const void* gaddr) {
    // GLOBAL_LOAD_ASYNC_TO_LDS_B128: per-lane 16B global -> LDS, tracked by ASYNCcnt
    asm volatile("global_load_async_to_lds_b128 %0, %1, off"
                 :: "v"(ldsoff), "v"(gaddr) : "memory");
}
__device__ __forceinline__ void async_b128_o(unsigned ldsoff, const void* gaddr, int ioff) {
    asm volatile("global_load_async_to_lds_b128 %0, %1, off offset:%2"
                 :: "v"(ldsoff), "v"(gaddr), "i"(ioff) : "memory");
}
__device__ __forceinline__ void wait_async0() {
    asm volatile("s_wait_asynccnt 0x0" ::: "memory");
}

__global__ __launch_bounds__(NTHREADS)
void gather_gemm_select_kernel(const int*   __restrict__ token,
                               const int*   __restrict__ need_mapper,
                               const float* __restrict__ emb,
                               const float* __restrict__ W,
                               const float* __restrict__ bias,
                               float*       __restrict__ out)
{
    __shared__ int   tokS[BM];
    __shared__ int   needS[BM];
    __shared__ float Asm[2][BM * ASTRIDE];   // gathered A tile, row-major [m][k]
    __shared__ float Bsm[2][BN * BTSTRIDE];  // W tile, TRANSPOSED  [n][k]

    const int t  = threadIdx.x;
    const int mb = blockIdx.x;           // M-tile index
    const int n0 = blockIdx.y * BN;      // N-tile base

    // ---- stage token ids + need flags for this 128-row tile ----
    if (t < BM) {
        int tk   = token[mb * BM + t];
        tokS[t]  = tk;
        needS[t] = need_mapper[tk];
    }
    __syncthreads();

    // ---- A: async DMA staging. 128 rows x 32 floats = 16KB/chunk.
    //      thread t copies 64 contiguous bytes of one gathered row.
    const int arow  = t >> 1;
    const int ahalf = (t & 1) * 16;                       // float offset in row
    const float* gA0 = emb + (size_t)tokS[arow] * DIM + ahalf;
    const unsigned ldsA[2] = { LDS_OFF(&Asm[0][arow * ASTRIDE + ahalf]),
                               LDS_OFF(&Asm[1][arow * ASTRIDE + ahalf]) };

    // ---- B: register staging (2 x float4 per thread), stored transposed.
    const int bk0 = t >> 4;              // k row of first float4 (0..15)
    const int bn0 = (t & 15) * 4;        // n col of first float4
    const int bk1 = bk0 + 16;            // second float4: k row +16, same n
    const float* gB0 = W + (size_t)bk0 * DIM + n0 + bn0;  // second: +16*DIM

    float4 br0, br1;

    auto stage_async_A = [&](int chunk, int buf) {
        const float* ga = gA0 + (size_t)chunk * KC;
        async_b128  (ldsA[buf],      ga);
        async_b128_o(ldsA[buf], ga, 16);
        async_b128_o(ldsA[buf], ga, 32);
        async_b128_o(ldsA[buf], ga, 48);
    };
    auto load_B = [&](int chunk) {
        const float* gb = gB0 + (size_t)chunk * KC * DIM;
        br0 = *(const float4*)(gb);
        br1 = *(const float4*)(gb + (size_t)16 * DIM);
    };
    auto store_B = [&](int buf) {
        float* bd = &Bsm[buf][0];
        bd[(bn0 + 0) * BTSTRIDE + bk0] = br0.x;
        bd[(bn0 + 1) * BTSTRIDE + bk0] = br0.y;
        bd[(bn0 + 2) * BTSTRIDE + bk0] = br0.z;
        bd[(bn0 + 3) * BTSTRIDE + bk0] = br0.w;
        bd[(bn0 + 0) * BTSTRIDE + bk1] = br1.x;
        bd[(bn0 + 1) * BTSTRIDE + bk1] = br1.y;
        bd[(bn0 + 2) * BTSTRIDE + bk1] = br1.z;
        bd[(bn0 + 3) * BTSTRIDE + bk1] = br1.w;
    };

    // ---- per-wave compute assignment ----
    const int lane = t & 31;
    const int wv   = t >> 5;         // 8 waves -> row groups of 16
    const int lm   = lane & 15;
    const int h    = lane >> 4;

    // A frag: row = wv*16 + lm, K = 4kk + 2h + v  (one aligned v2f)
    const int aoff = (wv * 16 + lm) * ASTRIDE + 2 * h;
    // B frag: n = nt*16 + lm, K = 4kk + 2h + v    (one aligned v2f, transposed LDS)
    const int boff = lm * BTSTRIDE + 2 * h;

    v8f acc[4] = {};

    // ---- prologue: stage chunk 0 ----
    stage_async_A(0, 0);
    load_B(0);
    store_B(0);

    // ---- main K loop: double-buffered, one wait+barrier per chunk ----
    for (int c = 0; c < NCHUNK; ++c) {
        const int buf = c & 1;
        wait_async0();                 // my async A writes into buf[c] landed
        __syncthreads();               // everyone's writes (A async + B ds) published
        if (c + 1 < NCHUNK) {
            stage_async_A(c + 1, buf ^ 1);
            load_B(c + 1);
        }

        const float* Ab = &Asm[buf][aoff];
        const float* Bb = &Bsm[buf][boff];
#pragma unroll
        for (int kk = 0; kk < KC / 4; ++kk) {
            v2f a = *(const v2f*)(Ab + 4 * kk);
#pragma unroll
            for (int nt = 0; nt < 4; ++nt) {
                v2f b = *(const v2f*)(Bb + nt * 16 * BTSTRIDE + 4 * kk);
                acc[nt] = __builtin_amdgcn_wmma_f32_16x16x4_f32(
                    /*neg_a=*/false, a, /*neg_b=*/false, b,
                    /*c_mod=*/(short)0, acc[nt],
                    /*reuse_a=*/false, /*reuse_b=*/false);
            }
        }
        if (c + 1 < NCHUNK) store_B(buf ^ 1);
    }

    // ---- epilogue: bias add + conditional select ----
#pragma unroll
    for (int nt = 0; nt < 4; ++nt) {
        const int   n  = n0 + nt * 16 + lm;
        const float bv = bias[n];
#pragma unroll
        for (int v = 0; v < 8; ++v) {
            const int m  = wv * 16 + v + 8 * h;   // C/D layout: M = v + 8*(lane>=16)
            const int gm = mb * BM + m;
            float r;
            if (needS[m]) r = acc[nt][v] + bv;
            else          r = emb[(size_t)tokS[m] * DIM + n];
            out[(size_t)gm * DIM + n] = r;
        }
    }
}

extern "C" void kernel_launch(void* const* d_in, const int* in_sizes, int n_in,
                              void* d_out, int out_size, void* d_ws, size_t ws_size,
                              hipStream_t stream) {
    const int*   token = (const int*)d_in[0];    // [32,512]
    const int*   need  = (const int*)d_in[1];    // [50000]
    const float* emb   = (const float*)d_in[2];  // [50000,768]
    const float* W     = (const float*)d_in[3];  // [768,768]
    const float* bias  = (const float*)d_in[4];  // [768]
    float*       out   = (float*)d_out;          // [32,512,768]

    const int ntok = in_sizes[0];                // 16384, multiple of BM
    dim3 grid(ntok / BM, DIM / BN);              // (128, 12)
    gather_gemm_select_kernel<<<grid, NTHREADS, 0, stream>>>(token, need, emb, W, bias, out);
}